// LutLayer_63780264346273
// MI455X (gfx1250) — compile-verified
//
#include <hip/hip_runtime.h>
#include <hip/hip_bf16.h>

// LUT-layer for MI455X (gfx1250, wave32).
//
// out[b,n] = sum_{t=0..63} sigmoid(lut[n,t]) * prod_{j=0..5} (bit_{5-j}(t) ? relu(1-x_j)+eps
//                                                                         : relu(x_j)+eps)
//
// t = (u<<2) | k  (u = bits5..2, k = bits1..0).  Per n:
//   R[u,b] = sum_k W[u,k] * G[k,b]   with W[u,k] = sigmoid(lut[n,4u+k])   <-- ONE wmma_f32_16x16x4_f32
//   out[b] = sum_u H[b,u] * R[u,b]   done as a per-lane 3-level fma tree + cross-half shuffle,
// because D's column b lives on lane pair (b, b+16): lane b holds u=0..7, lane b+16 holds u=8..15.
//
// Staging: [16b x 8n x 6] input tile -> LDS via coalesced 128-bit transfers, double buffered.
// Preferred path: CDNA5 GLOBAL_LOAD_ASYNC_TO_LDS_B128 (ASYNCcnt), fallback: reg-staged b128.

typedef __attribute__((ext_vector_type(2))) float v2f;
typedef __attribute__((ext_vector_type(8))) float v8f;
typedef int v4i_vec __attribute__((vector_size(16)));   // builtin's element type (int4 vector)
typedef __attribute__((address_space(1))) v4i_vec* g4p; // global int4*
typedef __attribute__((address_space(3))) v4i_vec* l4p; // LDS int4*

#if defined(__has_builtin)
# if __has_builtin(__builtin_amdgcn_global_load_async_to_lds_b128)
#  define USE_ASYNC_LDS 1
# endif
#endif

__device__ __forceinline__ void stage_b128(const float* __restrict__ src, float* dst) {
#if defined(USE_ASYNC_LDS)
    // DMA 16B global -> LDS, tracked by ASYNCcnt (no VGPR data, no loadcnt)
    __builtin_amdgcn_global_load_async_to_lds_b128((g4p)src, (l4p)dst, 0, 0);
#else
    *reinterpret_cast<float4*>(dst) = *reinterpret_cast<const float4*>(src);
#endif
}

__device__ __forceinline__ void stage_wait() {
#if defined(USE_ASYNC_LDS)
# if __has_builtin(__builtin_amdgcn_s_wait_asynccnt)
    __builtin_amdgcn_s_wait_asynccnt(0);
# else
    asm volatile("s_wait_asynccnt 0" ::: "memory");
# endif
#endif
}

#define BQ      1024          // batch
#define NQ      2048          // columns
#define NT      8             // n's per workgroup (one per wave, 8 waves = 256 threads)
#define BCHUNK  16            // batch elements per WMMA step
#define NCHUNKS 16            // chunks per workgroup
#define BTILE   (BCHUNK * NCHUNKS)   // 256 b's per workgroup
#define ROWF    52            // padded LDS floats per b-row (48 data + 4 pad -> conflict-free, 16B aligned)

__global__ __launch_bounds__(256)
void lut_layer_wmma(const float* __restrict__ in,   // [B, N, 6] fp32
                    const float* __restrict__ lut,  // [N, 64]   fp32
                    float* __restrict__ out)        // [B, N]    fp32
{
    __shared__ float sbuf[2][BCHUNK * ROWF];        // 2 x 3328 B, double-buffered tile

    const int wg = blockIdx.x;
    const int nt = wg & (NQ / NT - 1);              // 0..255
    const int bt = wg >> 8;                         // 0..3
    const int n0 = nt * NT;
    const int b0 = bt * BTILE;

    const int tid  = threadIdx.x;
    const int lane = tid & 31;
    const int wv   = tid >> 5;                      // wave id -> which n
    const int bl   = lane & 15;                     // batch-in-chunk / matrix row u
    const int hi   = lane >> 4;                     // lane half
    const int n    = n0 + wv;

    // ---- loop-invariant WMMA A operand: W[u,k] = sigmoid(lut[n, 4u+k]) ----
    // A 16x4 layout: lane u (lanes 0-15) holds K=0,1; lane u+16 holds K=2,3.
    const int   ub = bl * 4 + hi * 2;
    const float l0 = lut[(size_t)n * 64 + ub];
    const float l1 = lut[(size_t)n * 64 + ub + 1];
    v2f A;
    A.x = 1.0f / (1.0f + __expf(-l0));
    A.y = 1.0f / (1.0f + __expf(-l1));

    const int br  = tid / 12;                       // staging: b-row within chunk
    const int seg = tid % 12;                       // 12 x float4 = 48 floats per row

    // ---- stage chunk 0 (coalesced 128-bit transfers) ----
    if (tid < 192) {
        stage_b128(in + (size_t)(b0 + br) * (NQ * 6) + n0 * 6 + seg * 4,
                   &sbuf[0][br * ROWF + seg * 4]);
    }
    stage_wait();
    __syncthreads();

    for (int c = 0; c < NCHUNKS; ++c) {
        const bool stager = (tid < 192) && (c + 1 < NCHUNKS);
        const float* gsrc =
            in + (size_t)(b0 + (c + 1) * BCHUNK + br) * (NQ * 6) + n0 * 6 + seg * 4;
        float* ldst = &sbuf[(c + 1) & 1][br * ROWF + seg * 4];

#if defined(USE_ASYNC_LDS)
        // fire-and-forget DMA of chunk c+1; overlaps with all compute below
        if (stager) stage_b128(gsrc, ldst);
#else
        float4 nv;
        if (stager) nv = *reinterpret_cast<const float4*>(gsrc);
#endif

        // ---- compute chunk c: 16 outputs per wave ----
        const float* xp  = &sbuf[c & 1][bl * ROWF + wv * 6];
        const float2 x01 = *reinterpret_cast<const float2*>(xp);
        const float2 x23 = *reinterpret_cast<const float2*>(xp + 2);
        const float2 x45 = *reinterpret_cast<const float2*>(xp + 4);

        const float eps = 1e-7f;
        const float p0 = fmaxf(x01.x, 0.f) + eps, q0 = fmaxf(1.f - x01.x, 0.f) + eps;
        const float p1 = fmaxf(x01.y, 0.f) + eps, q1 = fmaxf(1.f - x01.y, 0.f) + eps;
        const float p2 = fmaxf(x23.x, 0.f) + eps, q2 = fmaxf(1.f - x23.x, 0.f) + eps;
        const float p3 = fmaxf(x23.y, 0.f) + eps, q3 = fmaxf(1.f - x23.y, 0.f) + eps;
        const float p4 = fmaxf(x45.x, 0.f) + eps, q4 = fmaxf(1.f - x45.x, 0.f) + eps;
        const float p5 = fmaxf(x45.y, 0.f) + eps, q5 = fmaxf(1.f - x45.y, 0.f) + eps;

        // B 4x16 operand: column b on lane pair; lane supplies G[2*hi + {0,1}, b]
        const float f4 = hi ? q4 : p4;              // k bit1 <-> factor j=4
        v2f Bv;
        Bv.x = f4 * p5;                             // k bit0 = 0 -> p5
        Bv.y = f4 * q5;                             // k bit0 = 1 -> q5

        v8f cz = {};
        // R[u,b] = sum_k W[u,k] * G[k,b]  -- fully-utilized 16x16x4 f32 WMMA
        v8f d = __builtin_amdgcn_wmma_f32_16x16x4_f32(
            false, A, false, Bv, (short)0, cz, false, false);

        // Contract remaining u bits (per-lane): VGPR r = u&7, bits (j1,j2,j3)
        const float e0 = fmaf(d[1], q3, d[0] * p3);
        const float e1 = fmaf(d[3], q3, d[2] * p3);
        const float e2 = fmaf(d[5], q3, d[4] * p3);
        const float e3 = fmaf(d[7], q3, d[6] * p3);
        const float f0 = fmaf(e1, q2, e0 * p2);
        const float f1 = fmaf(e3, q2, e2 * p2);
        const float g  = fmaf(f1, q1, f0 * p1);
        const float h  = g * (hi ? q0 : p0);        // u bit3 <-> factor j=0

        // lane b (u=0..7 part) + lane b+16 (u=8..15 part)
        const float tot = h + __shfl_xor(h, 16, 32);

        if (hi == 0) {
            out[(size_t)(b0 + c * BCHUNK + bl) * NQ + n] = tot;
        }

#if !defined(USE_ASYNC_LDS)
        if (stager) *reinterpret_cast<float4*>(ldst) = nv;
#endif
        // drain the chunk-(c+1) DMA, then one barrier per iteration
        stage_wait();
        __syncthreads();
    }
}

extern "C" void kernel_launch(void* const* d_in, const int* in_sizes, int n_in,
                              void* d_out, int out_size, void* d_ws, size_t ws_size,
                              hipStream_t stream) {
    (void)in_sizes; (void)n_in; (void)d_ws; (void)ws_size; (void)out_size;
    const float* in  = (const float*)d_in[0];   // [1024, 2048, 6] fp32
    const float* lut = (const float*)d_in[1];   // [2048, 64] fp32
    float* out = (float*)d_out;                 // [1024, 2048] fp32

    const int grid = (NQ / NT) * (BQ / BTILE);  // 256 n-tiles x 4 b-tiles = 1024 blocks
    lut_layer_wmma<<<grid, 256, 0, stream>>>(in, lut, out);
}